// MySCNN_68736656605439
// MI455X (gfx1250) — compile-verified
//
#include <hip/hip_runtime.h>
#include <hip/hip_bf16.h>

typedef __attribute__((ext_vector_type(2))) float v2f;
typedef __attribute__((ext_vector_type(8))) float v8f;

#define NEG_SLOPE 0.01f
#define KCHEB 5
#define MMAX 60000
// buffer rows padded by 16 so WMMA edge tiles can over-read AND over-write safely
#define BUFELTS ((size_t)(MMAX + 16) * 64)

// ---------------------------------------------------------------- utility
__global__ void zero_buf(float* __restrict__ p, size_t n) {
    size_t t = (size_t)blockIdx.x * blockDim.x + threadIdx.x;
    if (t < n) p[t] = 0.0f;
}

// x (B=4, C=4, M) -> xt (M, 16)  node-major
__global__ void transpose_in(const float* __restrict__ x, float* __restrict__ xt, int M) {
    int t = blockIdx.x * blockDim.x + threadIdx.x;
    if (t >= M * 16) return;
    int m = t >> 4;
    int bc = t & 15;
    xt[t] = x[(size_t)bc * M + m];
}

// ---------------------------------------------------------------- SpMM
// Y[row, c] += vals[e] * Xin[col, c]   (node-major, BC = 1<<bcShift channels)
// each lane handles 4 consecutive channels of one edge: b128 gather + 4 fp atomics
__global__ void spmm_atomic(const int* __restrict__ rows, const int* __restrict__ cols,
                            const float* __restrict__ vals, const float* __restrict__ Xin,
                            float* __restrict__ Y, int nnz, int laneShift, int bcShift) {
    long long tid = (long long)blockIdx.x * blockDim.x + threadIdx.x;
    int e = (int)(tid >> laneShift);
    if (e >= nnz) return;
    int c4 = (int)(tid & ((1 << laneShift) - 1)) << 2;
    int col = cols[e];
    int row = rows[e];
    float v = vals[e];
    const float4 xv = *(const float4*)(Xin + ((size_t)col << bcShift) + c4);
    float* yp = Y + ((size_t)row << bcShift) + c4;
    unsafeAtomicAdd(yp + 0, v * xv.x);
    unsafeAtomicAdd(yp + 1, v * xv.y);
    unsafeAtomicAdd(yp + 2, v * xv.z);
    unsafeAtomicAdd(yp + 3, v * xv.w);
}

// ---------------------------------------------------------------- einsum (WMMA f32)
// h[m, b*Co+o]  (=|+=)  bias[o] + sum_{k,i} theta[o,i,kbase+k] * Xk[m, b*Cin+i]
// One wave per (16-row m-tile, batch b), grid-striding over tiles.
// Theta B-fragments preloaded into registers once per wave; MODE/LEAKY are
// compile-time so the writeback is 8 straight-line (RMW|store) ops.
// A (16x4 f32): lane<16 holds A[lane][kk+0..1], lane>=16 holds A[lane-16][kk+2..3]
// B (4x16 f32): lane<16 holds B[0..1][lane],   lane>=16 holds B[2..3][lane-16]
template <int CIN, int CO, int KCOUNT, int MODE /*0: h=bias+acc, 1: h+=acc*/, int LEAKY>
__global__ __launch_bounds__(32)
void einsum_wmma(float* __restrict__ h, const float* __restrict__ X0,
                 const float* __restrict__ X1, size_t kstride, int kbase,
                 const float* __restrict__ theta, const float* __restrict__ bias,
                 int ntiles) {
    constexpr int BCin = 4 * CIN;
    constexpr int BCo  = 4 * CO;
    constexpr int NS   = CIN / 4;     // WMMA K-steps per Chebyshev term
    int lane = threadIdx.x;
    int hi = lane >> 4;               // selects K sub-pair / D row half
    int n  = lane & 15;               // A row within tile; B/D column

    // preload all B fragments (invariant across tiles)
    v2f bf[KCOUNT * NS];
#pragma unroll
    for (int k = 0; k < KCOUNT; ++k) {
#pragma unroll
        for (int s = 0; s < NS; ++s) {
            v2f f;
            if (n < CO) {
                const float* tp =
                    theta + ((size_t)n * CIN + (s * 4 + hi * 2)) * KCHEB + (kbase + k);
                f.x = tp[0];
                f.y = tp[KCHEB];
            } else {
                f.x = 0.0f;
                f.y = 0.0f;
            }
            bf[k * NS + s] = f;
        }
    }
    float bv = (n < CO) ? bias[n] : 0.0f;

    for (int tile = blockIdx.x; tile < ntiles; tile += gridDim.x) {
        int b  = tile & 3;
        int m0 = (tile >> 2) << 4;
        int rowA = m0 + n;
        v8f acc = {};
#pragma unroll
        for (int k = 0; k < KCOUNT; ++k) {
            const float* Xk = (k == 0) ? X0 : (X1 + (size_t)(k - 1) * kstride);
            const float* ap = Xk + (size_t)rowA * BCin + b * CIN + hi * 2;
#pragma unroll
            for (int s = 0; s < NS; ++s) {
                v2f a = *(const v2f*)(ap + s * 4);
                acc = __builtin_amdgcn_wmma_f32_16x16x4_f32(
                    false, a, false, bf[k * NS + s], (short)0, acc, false, false);
            }
        }
        // D element j -> row m0 + j + 8*hi, col n (padded buffer: no row guard)
        if (n < CO) {
            float* hp = h + (size_t)(m0 + 8 * hi) * BCo + b * CO + n;
#pragma unroll
            for (int j = 0; j < 8; ++j) {
                float v = acc[j] + (MODE ? hp[(size_t)j * BCo] : bv);
                if (LEAKY) v = (v >= 0.0f) ? v : NEG_SLOPE * v;
                hp[(size_t)j * BCo] = v;
            }
        }
    }
}

// incremental-path dispatcher (k / leaky known at launch time)
template <int CIN, int CO>
static inline void einsum_inc(unsigned eg, hipStream_t stream, float* h, const float* src,
                              int k, int leaky, const float* theta, const float* bias,
                              int ntiles) {
    if (k == 0)
        einsum_wmma<CIN, CO, 1, 0, 0><<<eg, 32, 0, stream>>>(h, src, src, 0, k, theta,
                                                             bias, ntiles);
    else if (leaky)
        einsum_wmma<CIN, CO, 1, 1, 1><<<eg, 32, 0, stream>>>(h, src, src, 0, k, theta,
                                                             bias, ntiles);
    else
        einsum_wmma<CIN, CO, 1, 1, 0><<<eg, 32, 0, stream>>>(h, src, src, 0, k, theta,
                                                             bias, ntiles);
}

// ---------------------------------------------------------------- finalize
// h3 (M,16) -> out (B,4,M) reference layout; block-reduced channel sums -> pooled
__global__ void finalize_out(const float* __restrict__ h3, float* __restrict__ out,
                             float* __restrict__ pooled, int M, int d) {
    __shared__ float red[256];
    int bc = threadIdx.x & 15;        // b*4+c
    int mi = threadIdx.x >> 4;        // 0..15
    float s = 0.0f;
    for (int m = blockIdx.x * 16 + mi; m < M; m += gridDim.x * 16) {
        float v = h3[(size_t)m * 16 + bc];
        out[(size_t)bc * M + m] = v;
        s += v;
    }
    red[threadIdx.x] = s;
    __syncthreads();
    for (int step = 8; step >= 1; step >>= 1) {
        if (mi < step) red[mi * 16 + bc] += red[(mi + step) * 16 + bc];
        __syncthreads();
    }
    if (mi == 0) {
        int b = bc >> 2, c = bc & 3;
        unsafeAtomicAdd(&pooled[b * 12 + d * 4 + c], red[bc]);
    }
}

// logits[b,j] = fc_b[j] + sum_ch fc_w[j,ch] * pooled[b,ch]
__global__ void fc_kernel(const float* __restrict__ pooled, const float* __restrict__ w,
                          const float* __restrict__ bias, float* __restrict__ logits) {
    int t = threadIdx.x;
    if (t >= 8) return;
    int b = t >> 1, j = t & 1;
    float s = bias[j];
    for (int ch = 0; ch < 12; ++ch) s += w[j * 12 + ch] * pooled[b * 12 + ch];
    logits[b * 2 + j] = s;
}

// ---------------------------------------------------------------- host
static inline unsigned cdiv(size_t a, unsigned b) { return (unsigned)((a + b - 1) / b); }

extern "C" void kernel_launch(void* const* d_in, const int* in_sizes, int n_in,
                              void* d_out, int out_size, void* d_ws, size_t ws_size,
                              hipStream_t stream) {
    (void)n_in; (void)out_size;
    float* ws = (float*)d_ws;
    float* pooled = ws;                               // 48 floats
    float* reg[6];
    for (int i = 0; i < 6; ++i) reg[i] = ws + 64 + (size_t)i * BUFELTS;

    const bool full = ws_size >= (64 + 6 * BUFELTS) * sizeof(float);

    zero_buf<<<1, 64, 0, stream>>>(pooled, 48);

    const int Cins[3] = {4, 16, 16};

    size_t outOff = 8;  // logits occupy d_out[0..8)
    for (int d = 0; d < 3; ++d) {
        const int*   rows = (const int*)d_in[4 * d + 0];
        const int*   cols = (const int*)d_in[4 * d + 1];
        const float* vals = (const float*)d_in[4 * d + 2];
        const float* x    = (const float*)d_in[4 * d + 3];
        int nnz = in_sizes[4 * d + 0];
        int M   = in_sizes[4 * d + 3] / 16;           // x is (4,4,M)
        int ntiles = ((M + 15) / 16) * 4;
        unsigned eg = (unsigned)((ntiles < 4096) ? ntiles : 4096);

        if (full) {
            // ---------- full-K path: X^0..X^4 all live, single einsum per layer
            int x0 = 0, hb = 5;                       // X^1..X^4 fixed in reg[1..4]
            transpose_in<<<cdiv((size_t)M * 16, 256), 256, 0, stream>>>(x, reg[x0], M);
            for (int l = 0; l < 3; ++l) {
                int BCin = 4 * Cins[l];
                int bcs = (BCin == 16) ? 4 : 6;       // log2(BC)
                int lsh = bcs - 2;                    // log2(lanes/edge), 4 ch per lane
                const float* theta = (const float*)d_in[12 + d * 6 + l * 2];
                const float* bias  = (const float*)d_in[12 + d * 6 + l * 2 + 1];
                for (int k = 1; k < KCHEB; ++k) {
                    const float* src = (k == 1) ? reg[x0] : reg[k - 1];
                    zero_buf<<<cdiv((size_t)M << bcs, 256), 256, 0, stream>>>(reg[k],
                                                                (size_t)M << bcs);
                    spmm_atomic<<<cdiv((size_t)nnz << lsh, 256), 256, 0, stream>>>(
                        rows, cols, vals, src, reg[k], nnz, lsh, bcs);
                }
                if (l == 0)
                    einsum_wmma<4, 16, 5, 0, 1><<<eg, 32, 0, stream>>>(reg[hb], reg[x0],
                        reg[1], BUFELTS, 0, theta, bias, ntiles);
                else if (l == 1)
                    einsum_wmma<16, 16, 5, 0, 1><<<eg, 32, 0, stream>>>(reg[hb], reg[x0],
                        reg[1], BUFELTS, 0, theta, bias, ntiles);
                else
                    einsum_wmma<16, 4, 5, 0, 0><<<eg, 32, 0, stream>>>(reg[hb], reg[x0],
                        reg[1], BUFELTS, 0, theta, bias, ntiles);
                int t = x0; x0 = hb; hb = t;          // h becomes next layer's X^0
            }
            finalize_out<<<256, 256, 0, stream>>>(reg[x0], (float*)d_out + outOff,
                                                  pooled, M, d);
        } else {
            // ---------- incremental path: 3 buffers, einsum accumulates per k
            int in = 0, aux = 1, hb = 2;
            transpose_in<<<cdiv((size_t)M * 16, 256), 256, 0, stream>>>(x, reg[in], M);
            for (int l = 0; l < 3; ++l) {
                int BCin = 4 * Cins[l];
                int bcs = (BCin == 16) ? 4 : 6;
                int lsh = bcs - 2;
                const float* theta = (const float*)d_in[12 + d * 6 + l * 2];
                const float* bias  = (const float*)d_in[12 + d * 6 + l * 2 + 1];
                int cur = in, nxt = aux;
                for (int k = 0; k < KCHEB; ++k) {
                    const float* src;
                    if (k == 0) {
                        src = reg[in];
                    } else {
                        zero_buf<<<cdiv((size_t)M << bcs, 256), 256, 0, stream>>>(reg[nxt],
                                                                    (size_t)M << bcs);
                        spmm_atomic<<<cdiv((size_t)nnz << lsh, 256), 256, 0, stream>>>(
                            rows, cols, vals, reg[cur], reg[nxt], nnz, lsh, bcs);
                        src = reg[nxt];
                        int t = cur; cur = nxt; nxt = t;
                    }
                    int leaky = (l < 2 && k == KCHEB - 1) ? 1 : 0;
                    if (l == 0)
                        einsum_inc<4, 16>(eg, stream, reg[hb], src, k, leaky, theta, bias,
                                          ntiles);
                    else if (l == 1)
                        einsum_inc<16, 16>(eg, stream, reg[hb], src, k, leaky, theta, bias,
                                           ntiles);
                    else
                        einsum_inc<16, 4>(eg, stream, reg[hb], src, k, leaky, theta, bias,
                                          ntiles);
                }
                int t = in; in = hb; hb = t;
            }
            finalize_out<<<256, 256, 0, stream>>>(reg[in], (float*)d_out + outOff,
                                                  pooled, M, d);
        }
        outOff += (size_t)16 * M;
    }

    fc_kernel<<<1, 32, 0, stream>>>(pooled, (const float*)d_in[30], (const float*)d_in[31],
                                    (float*)d_out);
}